// DataEmbedding_56470230007867
// MI455X (gfx1250) — compile-verified
//
#include <hip/hip_runtime.h>

typedef float v2f __attribute__((ext_vector_type(2)));
typedef float v4f __attribute__((ext_vector_type(4)));
typedef float v8f __attribute__((ext_vector_type(8)));

#define DMODEL 256
#define TSTEPS 12

// ---------------------------------------------------------------------------
// 1) init: ve = 0, deg = 1.0 (self-loop weight pre-seeded)
// ---------------------------------------------------------------------------
__global__ __launch_bounds__(256) void init_kernel(float* __restrict__ ve,
                                                   float* __restrict__ deg,
                                                   int veN, int N) {
    int i = blockIdx.x * 256 + threadIdx.x;
    if (i < veN)            ve[i] = 0.0f;
    else if (i < veN + N)   deg[i - veN] = 1.0f;
}

// ---------------------------------------------------------------------------
// 2) positional encoding: pe[t, 2i] = sin(t*div_i), pe[t, 2i+1] = cos(t*div_i)
//    div_i = exp(2i * (-ln(10000)/d)) ; tiny, one block
// ---------------------------------------------------------------------------
__global__ __launch_bounds__(256) void pe_kernel(float* __restrict__ pe) {
    const float c = -9.210340371976184f / (float)DMODEL;  // -ln(10000)/d
    for (int p = threadIdx.x; p < TSTEPS * (DMODEL / 2); p += 256) {
        int t = p / (DMODEL / 2);
        int i = p % (DMODEL / 2);
        float div = expf((2.0f * (float)i) * c);
        float ang = (float)t * div;
        pe[t * DMODEL + 2 * i]     = sinf(ang);
        pe[t * DMODEL + 2 * i + 1] = cosf(ang);
    }
}

// ---------------------------------------------------------------------------
// 3) degree: deg[col] += w  (self-loop "+1" already seeded by init)
// ---------------------------------------------------------------------------
__global__ __launch_bounds__(256) void deg_kernel(const int* __restrict__ ei,
                                                  const float* __restrict__ ew,
                                                  float* __restrict__ deg, int E) {
    int e = blockIdx.x * 256 + threadIdx.x;
    if (e >= E) return;
    unsafeAtomicAdd(&deg[ei[E + e]], ew[e]);   // col = ei[1][e]
}

// ---------------------------------------------------------------------------
// 4) dinv = deg > 0 ? rsqrt(deg) : 0   (in place)
// ---------------------------------------------------------------------------
__global__ __launch_bounds__(256) void dinv_kernel(float* __restrict__ deg, int N) {
    int i = blockIdx.x * 256 + threadIdx.x;
    if (i >= N) return;
    float d = deg[i];
    deg[i] = (d > 0.0f) ? rsqrtf(d) : 0.0f;
}

// ---------------------------------------------------------------------------
// 5) hg = emb_table @ W_gcn via V_WMMA_F32_16X16X4_F32.
//    One wave = one 16x16 output tile. K loop: 256 / 4 = 64 WMMA issues.
//    A frag (16x4 f32): lane l: m = l%16, K pair base = 2*(l/16)
//    B frag (4x16 f32): lane l: n = l%16, same K pair mapping
//    C/D (16x16 f32):   vgpr j: M = j + 8*(l/16), N = l%16
//    Loads clamp the row index so EXEC stays all-1s through every WMMA;
//    only the final stores are predicated (N=5000 is not a multiple of 16).
// ---------------------------------------------------------------------------
__global__ __launch_bounds__(32) void gcn_gemm_wmma(const float* __restrict__ emb,
                                                    const float* __restrict__ Wg,
                                                    float* __restrict__ hg, int N) {
    const int tileN = blockIdx.x & 15;       // 256 / 16 = 16 column tiles
    const int tileM = blockIdx.x >> 4;
    const int lane  = threadIdx.x;
    const int half  = lane >> 4;             // 0 | 1
    const int l16   = lane & 15;
    int m  = tileM * 16 + l16;
    int mc = (m < N) ? m : (N - 1);          // clamp (keeps EXEC uniform)
    const int n  = tileN * 16 + l16;
    const int kb = half * 2;

    const float* arow = emb + (size_t)mc * DMODEL;
    v8f acc = {};
    for (int kk = 0; kk < DMODEL; kk += 4) {
        v2f a, b;
        a.x = arow[kk + kb];
        a.y = arow[kk + kb + 1];
        b.x = Wg[(size_t)(kk + kb)     * DMODEL + n];
        b.y = Wg[(size_t)(kk + kb + 1) * DMODEL + n];
        acc = __builtin_amdgcn_wmma_f32_16x16x4_f32(
                  /*neg_a=*/false, a, /*neg_b=*/false, b,
                  /*c_mod=*/(short)0, acc, /*reuse_a=*/false, /*reuse_b=*/false);
    }

    const int mbase = tileM * 16 + half * 8;
#pragma unroll
    for (int j = 0; j < 8; ++j) {
        int mm = mbase + j;
        if (mm < N) hg[(size_t)mm * DMODEL + n] = acc[j];
    }
}

// ---------------------------------------------------------------------------
// 6) scatter: ve[col] += dinv[row]*w*dinv[col] * hg[row]
//    one block per (edge | self-loop), 256 lanes = 256 channels
// ---------------------------------------------------------------------------
__global__ __launch_bounds__(256) void gcn_scatter(const int* __restrict__ ei,
                                                   const float* __restrict__ ew,
                                                   const float* __restrict__ dinv,
                                                   const float* __restrict__ hg,
                                                   float* __restrict__ ve,
                                                   int N, int E) {
    int e = blockIdx.x;
    int d = threadIdx.x;
    int r, c; float w;
    if (e < E) { r = ei[e]; c = ei[E + e]; w = ew[e]; }
    else       { r = c = e - E; w = 1.0f; }          // self loops
    float norm = dinv[r] * w * dinv[c];
    float val  = norm * hg[(size_t)r * DMODEL + d];
    unsafeAtomicAdd(&ve[(size_t)c * DMODEL + d], val);
}

// ---------------------------------------------------------------------------
// 7) fused output: out[row, j..j+3] = x[row,:]@W_lin + b_lin + b_gcn + pe[t] + ve[n]
//    row = ((b*N)+n)*T + t ; 4 channels per thread, NT 128-bit stores.
// ---------------------------------------------------------------------------
__global__ __launch_bounds__(256) void fused_out(const float* __restrict__ x,
                                                 const float* __restrict__ Wl,
                                                 const float* __restrict__ bl,
                                                 const float* __restrict__ bg,
                                                 const float* __restrict__ pe,
                                                 const float* __restrict__ ve,
                                                 float* __restrict__ out,
                                                 int nRows, int N) {
    long long gid = (long long)blockIdx.x * 256 + threadIdx.x;
    int row = (int)(gid >> 6);                 // 64 threads per output row
    if (row >= nRows) return;
    int j = ((int)gid & 63) * 4;
    int t = row % TSTEPS;
    int n = (row / TSTEPS) % N;

    const float* xp = x + (size_t)row * 3;
    float x0 = xp[0], x1 = xp[1], x2 = xp[2];

    v4f w0 = *(const v4f*)(Wl + j);
    v4f w1 = *(const v4f*)(Wl + DMODEL + j);
    v4f w2 = *(const v4f*)(Wl + 2 * DMODEL + j);
    v4f bb = *(const v4f*)(bl + j);
    v4f bg4 = *(const v4f*)(bg + j);
    v4f pe4 = *(const v4f*)(pe + t * DMODEL + j);
    v4f ve4 = *(const v4f*)(ve + (size_t)n * DMODEL + j);

    v4f base = bb + bg4 + pe4 + ve4;
    v4f r;
    r.x = fmaf(x0, w0.x, fmaf(x1, w1.x, fmaf(x2, w2.x, base.x)));
    r.y = fmaf(x0, w0.y, fmaf(x1, w1.y, fmaf(x2, w2.y, base.y)));
    r.z = fmaf(x0, w0.z, fmaf(x1, w1.z, fmaf(x2, w2.z, base.z)));
    r.w = fmaf(x0, w0.w, fmaf(x1, w1.w, fmaf(x2, w2.w, base.w)));

    __builtin_nontemporal_store(r, (v4f*)(out + (size_t)row * DMODEL + j));
}

// ---------------------------------------------------------------------------
extern "C" void kernel_launch(void* const* d_in, const int* in_sizes, int n_in,
                              void* d_out, int out_size, void* d_ws, size_t ws_size,
                              hipStream_t stream) {
    const float* x   = (const float*)d_in[0];
    // d_in[1] = x_mark (unused by forward)
    const int*   ei  = (const int*)d_in[2];
    const float* ew  = (const float*)d_in[3];
    const float* Wl  = (const float*)d_in[4];
    const float* bl  = (const float*)d_in[5];
    const float* emb = (const float*)d_in[6];
    const float* Wg  = (const float*)d_in[7];
    const float* bg  = (const float*)d_in[8];

    const int E     = in_sizes[3];             // 160000
    const int N     = in_sizes[6] / DMODEL;    // 5000
    const int nRows = in_sizes[0] / 3;         // B*N*T = 480000

    // workspace layout (floats): ve | hg | deg/dinv | pe   (~10.3 MB total)
    float* ws  = (float*)d_ws;
    float* ve  = ws;
    float* hg  = ve  + (size_t)N * DMODEL;
    float* deg = hg  + (size_t)N * DMODEL;
    float* pe  = deg + N;

    const int veN = N * DMODEL;
    init_kernel<<<(veN + N + 255) / 256, 256, 0, stream>>>(ve, deg, veN, N);
    pe_kernel<<<1, 256, 0, stream>>>(pe);
    deg_kernel<<<(E + 255) / 256, 256, 0, stream>>>(ei, ew, deg, E);
    dinv_kernel<<<(N + 255) / 256, 256, 0, stream>>>(deg, N);

    const int mTiles = (N + 15) / 16;          // 313
    gcn_gemm_wmma<<<mTiles * 16, 32, 0, stream>>>(emb, Wg, hg, N);

    gcn_scatter<<<E + N, 256, 0, stream>>>(ei, ew, deg, hg, ve, N, E);

    long long tot = (long long)nRows * 64;     // 4 channels per thread
    fused_out<<<(unsigned)((tot + 255) / 256), 256, 0, stream>>>(
        x, Wl, bl, bg, pe, ve, (float*)d_out, nRows, N);
}